// LightGCN_74431783239693
// MI455X (gfx1250) — compile-verified
//
#include <hip/hip_runtime.h>
#include <hip/hip_bf16.h>
#include <math.h>

// ---------------- problem constants (match reference) ----------------
#define NUM_USERS 100000
#define NUM_ITEMS 50000
#define DIM       64
#define N_LAYERS  3
#define N_TOTAL   (NUM_USERS + 1 + NUM_ITEMS)   // 150001
#define N_EDGES   2000000
#define BATCH     8192

typedef float v2f __attribute__((ext_vector_type(2)));
typedef float v8f __attribute__((ext_vector_type(8)));

// Hardware fp32 atomic add (no return) — guarantees global_atomic_add_f32,
// not a CAS loop. Ordering vs. kernel end is handled by the implicit
// s_wait_idle before s_endpgm.
__device__ __forceinline__ void gatomic_add_f32(float* p, float v) {
    asm volatile("global_atomic_add_f32 %0, %1, off" :: "v"(p), "v"(v) : "memory");
}

// ---------------- kernel 1: build cur = concat(user_emb,item_emb); acc = cur
__global__ __launch_bounds__(256) void lgcn_init(const float4* __restrict__ ue,
                                                 const float4* __restrict__ ie,
                                                 float4* __restrict__ cur,
                                                 float4* __restrict__ acc,
                                                 int nvec, int uvec) {
    int i = blockIdx.x * 256 + threadIdx.x;
    if (i >= nvec) return;
    float4 v = (i < uvec) ? ue[i] : ie[i - uvec];
    cur[i] = v;
    acc[i] = v;
}

// ---------------- kernel 2: zero a float4 buffer
__global__ __launch_bounds__(256) void lgcn_zero4(float4* __restrict__ p, int nvec) {
    int i = blockIdx.x * 256 + threadIdx.x;
    if (i >= nvec) return;
    p[i] = make_float4(0.f, 0.f, 0.f, 0.f);
}

// ---------------- kernel 3: edge scatter (segment_sum of edge_val * cur[col])
// 16 lanes per edge, each lane owns a float4 (16B) slice of the 256B row.
__global__ __launch_bounds__(256) void lgcn_scatter(const int*   __restrict__ erow,
                                                    const int*   __restrict__ ecol,
                                                    const float* __restrict__ eval,
                                                    const float* __restrict__ cur,
                                                    float*       __restrict__ nxt) {
    int t = blockIdx.x * 256 + threadIdx.x;
    if (t >= N_EDGES * 16) return;
    int e = t >> 4;
    int l = t & 15;
    int src = ecol[e];
    int dst = erow[e];
    float w = eval[e];
    const float4 x = *reinterpret_cast<const float4*>(cur + (size_t)src * DIM + l * 4);
    float* o = nxt + (size_t)dst * DIM + l * 4;
    gatomic_add_f32(o + 0, w * x.x);
    gatomic_add_f32(o + 1, w * x.y);
    gatomic_add_f32(o + 2, w * x.z);
    gatomic_add_f32(o + 3, w * x.w);
}

// ---------------- kernel 4: acc += nxt
__global__ __launch_bounds__(256) void lgcn_accum(float4* __restrict__ acc,
                                                  const float4* __restrict__ nxt,
                                                  int nvec) {
    int i = blockIdx.x * 256 + threadIdx.x;
    if (i >= nvec) return;
    float4 a = acc[i];
    float4 b = nxt[i];
    a.x += b.x; a.y += b.y; a.z += b.z; a.w += b.w;
    acc[i] = a;
}

// ---------------- kernel 5: zero scalar output
__global__ void lgcn_zero_out(float* out) { out[0] = 0.f; }

// ---------------- kernel 6: batched dot via V_WMMA_F32_16X16X4_F32 + BCE loss
// Each wave32 computes 16 dot products: A = 16 u-rows x K, B = K x 16 v-rows,
// K accumulated 4 at a time over DIM=64; dots are the 16x16 tile diagonal.
// A layout (ISA 7.12.2): lane(0..15)->M=lane K={0,1}; lane(16..31)->M=lane-16 K={2,3}.
// C/D layout: VGPR r holds M=r (lanes 0-15) and M=r+8 (lanes 16-31), N=lane&15.
__global__ __launch_bounds__(256) void lgcn_loss(const int*   __restrict__ users,
                                                 const int*   __restrict__ items,
                                                 const float* __restrict__ labels,
                                                 const float* __restrict__ acc,
                                                 float*       __restrict__ out) {
    __shared__ float tile[8][16][16];   // one 16x16 f32 tile per wave
    __shared__ float partial[256];

    const int tid  = threadIdx.x;
    const int wave = tid >> 5;
    const int lane = tid & 31;
    const int b0   = (blockIdx.x * 8 + wave) * 16;   // 16 batch rows per wave
    const int row  = b0 + (lane & 15);

    const int u = users[row];
    const int it = items[row];
    const int khalf = (lane >> 4) << 1;              // 0 or 2: K-subcolumn pair
    const float* up = acc + (size_t)u * DIM + khalf;
    const float* vp = acc + (size_t)(NUM_USERS + 1 + it) * DIM + khalf;

    v8f c = {0.f, 0.f, 0.f, 0.f, 0.f, 0.f, 0.f, 0.f};
#pragma unroll
    for (int k = 0; k < DIM / 4; ++k) {
        v2f a = *reinterpret_cast<const v2f*>(up + 4 * k);
        v2f b = *reinterpret_cast<const v2f*>(vp + 4 * k);
        c = __builtin_amdgcn_wmma_f32_16x16x4_f32(
                /*neg_a=*/false, a, /*neg_b=*/false, b,
                /*c_mod=*/(short)0, c, /*reuse_a=*/false, /*reuse_b=*/false);
    }

    // Spill the C/D tile to LDS so we can read the diagonal.
    const int mhi = (lane >> 4) * 8;
    const int n   = lane & 15;
#pragma unroll
    for (int r = 0; r < 8; ++r) tile[wave][mhi + r][n] = c[r];
    __syncthreads();

    float term = 0.f;
    if (lane < 16) {
        // layer mean contributes 1/4 to u and 1/4 to v -> 1/16 on the dot
        float g = 0.0625f * tile[wave][lane][lane];
        float y = labels[b0 + lane];
        term = fmaxf(g, 0.f) - g * y + log1pf(expf(-fabsf(g)));
    }
    partial[tid] = term;
    __syncthreads();
#pragma unroll
    for (int s = 128; s > 0; s >>= 1) {
        if (tid < s) partial[tid] += partial[tid + s];
        __syncthreads();
    }
    if (tid == 0) gatomic_add_f32(out, partial[0] * (1.0f / BATCH));
}

// ---------------- host-side launch sequence ----------------
extern "C" void kernel_launch(void* const* d_in, const int* in_sizes, int n_in,
                              void* d_out, int out_size, void* d_ws, size_t ws_size,
                              hipStream_t stream) {
    (void)in_sizes; (void)n_in; (void)out_size; (void)ws_size;

    const int*   users   = (const int*)  d_in[0];
    const int*   items   = (const int*)  d_in[1];
    const float* labels  = (const float*)d_in[2];
    const int*   erow    = (const int*)  d_in[3];
    const int*   ecol    = (const int*)  d_in[4];
    const float* eval    = (const float*)d_in[5];
    const float* uemb    = (const float*)d_in[6];
    const float* iemb    = (const float*)d_in[7];
    float*       out     = (float*)d_out;

    const size_t embFloats = (size_t)N_TOTAL * DIM;          // 9,600,064
    float* cur = (float*)d_ws;
    float* nxt = cur + embFloats;
    float* acc = nxt + embFloats;

    const int nvec = (int)(embFloats / 4);                   // 2,400,016 float4s
    const int uvec = (NUM_USERS + 1) * DIM / 4;              // 1,600,016
    const int gridV = (nvec + 255) / 256;
    const int gridE = (N_EDGES * 16 + 255) / 256;

    lgcn_init<<<gridV, 256, 0, stream>>>((const float4*)uemb, (const float4*)iemb,
                                         (float4*)cur, (float4*)acc, nvec, uvec);

    for (int layer = 0; layer < N_LAYERS; ++layer) {
        lgcn_zero4  <<<gridV, 256, 0, stream>>>((float4*)nxt, nvec);
        lgcn_scatter<<<gridE, 256, 0, stream>>>(erow, ecol, eval, cur, nxt);
        lgcn_accum  <<<gridV, 256, 0, stream>>>((float4*)acc, (const float4*)nxt, nvec);
        float* t = cur; cur = nxt; nxt = t;
    }

    lgcn_zero_out<<<1, 1, 0, stream>>>(out);
    // BATCH / (8 waves * 16 rows) = 64 blocks
    lgcn_loss<<<BATCH / 128, 256, 0, stream>>>(users, items, labels, acc, out);
}